// VQ_88699664597022
// MI455X (gfx1250) — compile-verified
//
#include <hip/hip_runtime.h>

typedef float v2f __attribute__((ext_vector_type(2)));
typedef float v8f __attribute__((ext_vector_type(8)));

#define D          256
#define CODEBOOK   1024
#define NTOK       (16 * 4096)          // 65536 tokens
#define NELEM      ((size_t)NTOK * D)   // 16777216
#define NBLOCKS    (NTOK / 16)          // 4096

// ---------------------------------------------------------------------------
// Kernel 1: enorm[k] = sum_d e[k][d]^2   (one wave per code row)
// ---------------------------------------------------------------------------
__global__ __launch_bounds__(256) void vq_enorm_kernel(const float* __restrict__ emb,
                                                       float* __restrict__ enorm) {
  const int wave = threadIdx.x >> 5;
  const int lane = threadIdx.x & 31;
  const int row  = blockIdx.x * 8 + wave;
  const float* er = emb + (size_t)row * D + lane * 8;
  float4 a = *(const float4*)er;
  float4 b = *(const float4*)(er + 4);
  float s = a.x*a.x + a.y*a.y + a.z*a.z + a.w*a.w
          + b.x*b.x + b.y*b.y + b.z*b.z + b.w*b.w;
#pragma unroll
  for (int off = 16; off > 0; off >>= 1) s += __shfl_xor(s, off, 32);
  if (lane == 0) enorm[row] = s;
}

// ---------------------------------------------------------------------------
// Kernel 2: per-16-token block: WMMA f32 GEMM vs all 1024 codes, argmin,
//           gather codeword -> values_st, fused squared-error partial sum.
// ---------------------------------------------------------------------------
__global__ __launch_bounds__(256) void vq_main_kernel(const float* __restrict__ x,
                                                      const float* __restrict__ emb,
                                                      const float* __restrict__ enorm,
                                                      float* __restrict__ out_vals,
                                                      float* __restrict__ out_idx,
                                                      float* __restrict__ partials) {
  __shared__ float lds_x[16][260];      // padded: 260*4B row stride -> conflict-free
  __shared__ float s_score[8][16];
  __shared__ int   s_idx[8][16];
  __shared__ int   s_final[16];
  __shared__ float s_red[8];

  const int tid        = threadIdx.x;
  const int wave       = tid >> 5;
  const int lane       = tid & 31;
  const int lrow       = lane & 15;            // M (A) or N (B/C) index in tile
  const int khalf      = (lane >> 4) << 1;     // K pair base: 0 or 2
  const int token_base = blockIdx.x * 16;

  // ---- stage X tile [16 x 256] into LDS (coalesced) ----
  {
    const int r  = tid >> 4;           // 0..15
    const int c0 = (tid & 15) * 16;    // 0..240
    const float* xr = x + (size_t)(token_base + r) * D + c0;
#pragma unroll
    for (int j = 0; j < 16; j += 4)
      *(float4*)&lds_x[r][c0 + j] = *(const float4*)(xr + j);
  }
  __syncthreads();

  // ---- GEMM: this wave covers codes [wave*128, wave*128+128) ----
  const int code_base = wave * 128;
  const float* ebase  = emb + (size_t)(code_base + lrow) * D;

  v8f acc[8];
#pragma unroll
  for (int t = 0; t < 8; ++t) acc[t] = (v8f){0.f,0.f,0.f,0.f,0.f,0.f,0.f,0.f};

  for (int k0 = 0; k0 < D; k0 += 4) {
    v2f a = *(const v2f*)&lds_x[lrow][k0 + khalf];
#pragma unroll
    for (int t = 0; t < 8; ++t) {
      v2f b = *(const v2f*)(ebase + (size_t)t * 16 * D + k0 + khalf);
      acc[t] = __builtin_amdgcn_wmma_f32_16x16x4_f32(
          /*neg_a=*/false, a, /*neg_b=*/false, b,
          /*c_mod=*/(short)0, acc[t], /*reuse_a=*/false, /*reuse_b=*/false);
    }
  }

  // ---- score = |e|^2 - 2*dot ; per-lane min over the 8 N-tiles ----
  // acc[t][r] = dot(token m, code n): m = r + 8*(lane>>4), n = code_base + t*16 + lrow
  float bestS[8];
  int   bestI[8];
#pragma unroll
  for (int r = 0; r < 8; ++r) { bestS[r] = 3.4e38f; bestI[r] = 0x7fffffff; }
#pragma unroll
  for (int t = 0; t < 8; ++t) {
    const int n = code_base + t * 16 + lrow;
    const float en = enorm[n];
#pragma unroll
    for (int r = 0; r < 8; ++r) {
      const float s = en - 2.0f * acc[t][r];
      if (s < bestS[r] || (s == bestS[r] && n < bestI[r])) { bestS[r] = s; bestI[r] = n; }
    }
  }

  // ---- butterfly min within each 16-lane half (same token m per r) ----
#pragma unroll
  for (int r = 0; r < 8; ++r) {
    float s = bestS[r];
    int   i = bestI[r];
#pragma unroll
    for (int off = 1; off < 16; off <<= 1) {
      float s2 = __shfl_xor(s, off, 32);
      int   i2 = __shfl_xor(i, off, 32);
      if (s2 < s || (s2 == s && i2 < i)) { s = s2; i = i2; }
    }
    if (lrow == 0) {
      const int m = r + ((lane >> 4) << 3);
      s_score[wave][m] = s;
      s_idx[wave][m]   = i;
    }
  }
  __syncthreads();

  // ---- cross-wave reduce: final argmin per token ----
  if (tid < 16) {
    float s = s_score[0][tid];
    int   i = s_idx[0][tid];
#pragma unroll
    for (int w = 1; w < 8; ++w) {
      const float s2 = s_score[w][tid];
      const int   i2 = s_idx[w][tid];
      if (s2 < s || (s2 == s && i2 < i)) { s = s2; i = i2; }
    }
    s_final[tid] = i;
    out_idx[token_base + tid] = (float)i;   // int output stored as f32 per contract
  }
  __syncthreads();

  // ---- gather codeword -> values_st; fused (x - e)^2 partial ----
  const int m  = tid >> 4;
  const int c0 = (tid & 15) * 16;
  const float* er = emb + (size_t)s_final[m] * D + c0;
  const float* xr = x + (size_t)(token_base + m) * D + c0;
  float*       vr = out_vals + (size_t)(token_base + m) * D + c0;
  float lsum = 0.f;
#pragma unroll
  for (int j = 0; j < 16; j += 4) {
    float4 e4 = *(const float4*)(er + j);
    float4 x4 = *(const float4*)(xr + j);
    *(float4*)(vr + j) = e4;
    const float d0 = x4.x - e4.x, d1 = x4.y - e4.y;
    const float d2 = x4.z - e4.z, d3 = x4.w - e4.w;
    lsum += d0 * d0 + d1 * d1 + d2 * d2 + d3 * d3;
  }
#pragma unroll
  for (int off = 16; off > 0; off >>= 1) lsum += __shfl_xor(lsum, off, 32);
  if (lane == 0) s_red[wave] = lsum;
  __syncthreads();
  if (tid == 0) {
    float t = 0.f;
#pragma unroll
    for (int w = 0; w < 8; ++w) t += s_red[w];
    partials[blockIdx.x] = t;
  }
}

// ---------------------------------------------------------------------------
// Kernel 3: loss = (1 + beta) * sum(partials) / NELEM   (fp64 accumulate)
// ---------------------------------------------------------------------------
__global__ __launch_bounds__(256) void vq_finalize_kernel(const float* __restrict__ partials,
                                                          float* __restrict__ loss_out) {
  __shared__ double sd[8];
  double s = 0.0;
  for (int i = threadIdx.x; i < NBLOCKS; i += 256) s += (double)partials[i];
#pragma unroll
  for (int off = 16; off > 0; off >>= 1) s += __shfl_xor(s, off, 32);
  const int wave = threadIdx.x >> 5;
  const int lane = threadIdx.x & 31;
  if (lane == 0) sd[wave] = s;
  __syncthreads();
  if (threadIdx.x == 0) {
    double t = 0.0;
#pragma unroll
    for (int w = 0; w < 8; ++w) t += sd[w];
    loss_out[0] = (float)(1.1 * t / (double)NELEM);
  }
}

// ---------------------------------------------------------------------------
extern "C" void kernel_launch(void* const* d_in, const int* in_sizes, int n_in,
                              void* d_out, int out_size, void* d_ws, size_t ws_size,
                              hipStream_t stream) {
  const float* x   = (const float*)d_in[0];   // [16, 4096, 256]
  const float* emb = (const float*)d_in[1];   // [1024, 256]

  float* out_vals = (float*)d_out;            // 16777216 floats
  float* out_idx  = out_vals + NELEM;         // 65536 floats (indexes)
  float* loss_out = out_idx + NTOK;           // 1 float

  float* enorm    = (float*)d_ws;             // 1024 floats
  float* partials = enorm + CODEBOOK;         // 4096 floats

  vq_enorm_kernel<<<CODEBOOK / 8, 256, 0, stream>>>(emb, enorm);
  vq_main_kernel<<<NBLOCKS, 256, 0, stream>>>(x, emb, enorm,
                                              out_vals, out_idx, partials);
  vq_finalize_kernel<<<1, 256, 0, stream>>>(partials, loss_out);
}